// ICP_44444321579432
// MI455X (gfx1250) — compile-verified
//
#include <hip/hip_runtime.h>
#include <math.h>

typedef __attribute__((ext_vector_type(2))) float v2f;
typedef __attribute__((ext_vector_type(8))) float v8f;

#define NPTS   192
#define NTILE  12
#define MAXIT  200
#define TOLCNV 1e-3f

__device__ __forceinline__ float wave_sum(float v) {
#pragma unroll
  for (int m = 1; m < 32; m <<= 1) v += __shfl_xor(v, m, 32);
  return v;
}

// Kabsch: given H = sum (a-ca)(b-cb)^T, produce proper rotation R = V U^T and t = cb - R ca.
__device__ void kabsch3(const float H[3][3], const float ca[3], const float cb[3],
                        float R[3][3], float tv[3]) {
  // S = H^T H (symmetric PSD)
  float S[3][3];
#pragma unroll
  for (int r = 0; r < 3; ++r)
#pragma unroll
    for (int c = 0; c < 3; ++c)
      S[r][c] = H[0][r] * H[0][c] + H[1][r] * H[1][c] + H[2][r] * H[2][c];

  float V[3][3] = {{1.f, 0.f, 0.f}, {0.f, 1.f, 0.f}, {0.f, 0.f, 1.f}};
  const int P[3] = {0, 0, 1}, Q[3] = {1, 2, 2};
  for (int sweep = 0; sweep < 12; ++sweep) {
#pragma unroll
    for (int k = 0; k < 3; ++k) {
      const int p = P[k], q = Q[k];
      const float apq = S[p][q];
      if (fabsf(apq) < 1e-30f) continue;
      const float tau = (S[q][q] - S[p][p]) / (2.0f * apq);
      const float tt = (tau >= 0.0f ? 1.0f : -1.0f) / (fabsf(tau) + sqrtf(1.0f + tau * tau));
      const float cc = 1.0f / sqrtf(1.0f + tt * tt);
      const float ss = tt * cc;
#pragma unroll
      for (int i = 0; i < 3; ++i) {            // S = S * J
        const float sip = S[i][p], siq = S[i][q];
        S[i][p] = cc * sip - ss * siq;
        S[i][q] = ss * sip + cc * siq;
      }
#pragma unroll
      for (int i = 0; i < 3; ++i) {            // S = J^T * S
        const float spi = S[p][i], sqi = S[q][i];
        S[p][i] = cc * spi - ss * sqi;
        S[q][i] = ss * spi + cc * sqi;
      }
#pragma unroll
      for (int i = 0; i < 3; ++i) {            // V = V * J
        const float vip = V[i][p], viq = V[i][q];
        V[i][p] = cc * vip - ss * viq;
        V[i][q] = ss * vip + cc * viq;
      }
    }
  }

  float lam[3] = {S[0][0], S[1][1], S[2][2]};
  float detSign = 1.0f;  // det(V)=+1 from Jacobi; each column swap flips it.
#pragma unroll
  for (int a = 0; a < 2; ++a)
#pragma unroll
    for (int bb = 0; bb < 2; ++bb)
      if (bb < 2 - a && lam[bb] < lam[bb + 1]) {
        const float tl = lam[bb]; lam[bb] = lam[bb + 1]; lam[bb + 1] = tl;
#pragma unroll
        for (int i = 0; i < 3; ++i) {
          const float tvv = V[i][bb]; V[i][bb] = V[i][bb + 1]; V[i][bb + 1] = tvv;
        }
        detSign = -detSign;
      }

  // U columns: u0 = Hv0/|Hv0|, u1 = GS(Hv1), u2 = u0 x u1  (det(U)=+1)
  float u0[3], u1[3], u2[3], w[3];
#pragma unroll
  for (int r = 0; r < 3; ++r)
    w[r] = H[r][0] * V[0][0] + H[r][1] * V[1][0] + H[r][2] * V[2][0];
  {
    const float n = fmaxf(sqrtf(w[0] * w[0] + w[1] * w[1] + w[2] * w[2]), 1e-20f);
#pragma unroll
    for (int r = 0; r < 3; ++r) u0[r] = w[r] / n;
  }
#pragma unroll
  for (int r = 0; r < 3; ++r)
    w[r] = H[r][0] * V[0][1] + H[r][1] * V[1][1] + H[r][2] * V[2][1];
  {
    const float d = w[0] * u0[0] + w[1] * u0[1] + w[2] * u0[2];
#pragma unroll
    for (int r = 0; r < 3; ++r) w[r] -= d * u0[r];
    const float n = fmaxf(sqrtf(w[0] * w[0] + w[1] * w[1] + w[2] * w[2]), 1e-20f);
#pragma unroll
    for (int r = 0; r < 3; ++r) u1[r] = w[r] / n;
  }
  u2[0] = u0[1] * u1[2] - u0[2] * u1[1];
  u2[1] = u0[2] * u1[0] - u0[0] * u1[2];
  u2[2] = u0[0] * u1[1] - u0[1] * u1[0];

  if (detSign < 0.0f) {  // reflection fix: flip V column tied to smallest sigma
#pragma unroll
    for (int i = 0; i < 3; ++i) V[i][2] = -V[i][2];
  }

#pragma unroll
  for (int r = 0; r < 3; ++r)
#pragma unroll
    for (int c = 0; c < 3; ++c)
      R[r][c] = V[r][0] * u0[c] + V[r][1] * u1[c] + V[r][2] * u2[c];

#pragma unroll
  for (int r = 0; r < 3; ++r)
    tv[r] = cb[r] - (R[r][0] * ca[0] + R[r][1] * ca[1] + R[r][2] * ca[2]);
}

__global__ __launch_bounds__(NPTS) void icp_wmma_kernel(const float* __restrict__ xyzs,
                                                        float* __restrict__ out) {
  // Zero-padded 4-row SoA (rows: x, y, z, 0) -> branch-free WMMA fragment loads.
  __shared__ float sS4[4 * NPTS];                     // moving src, row 3 == 0
  __shared__ float sB4[4 * NPTS];                     // target B,   row 3 == 0
  __shared__ float sAx[NPTS], sAy[NPTS], sAz[NPTS];   // original A
  __shared__ float sSn[NPTS], sBn[NPTS];              // squared norms
  __shared__ int   sNN[NPTS];
  __shared__ float sND[NPTS];
  __shared__ float sPart[6][16];
  __shared__ float sRt[12];                           // R row-major + t
  __shared__ float sCtl[2];                           // prev_err, done

  const int tid  = threadIdx.x;
  const int lane = tid & 31;
  const int wid  = tid >> 5;
  const int half = (lane >> 4) & 1;
  const int l16  = lane & 15;
  const int bidx = blockIdx.x;
  const int krow = 2 * half * NPTS;   // fragment base row: K=0 (half 0) / K=2 (half 1)

  // Load + subsample: (3,19200) viewed as (80,240,3), take [::10, ::10].
  {
    const size_t base = (size_t)bidx * 2u * 57600u;
    const int i = tid / 24, j = tid % 24;
    const size_t off = (size_t)i * 7200 + (size_t)j * 30;
    const float ax = xyzs[base + off + 0];
    const float ay = xyzs[base + off + 1];
    const float az = xyzs[base + off + 2];
    const float bx = xyzs[base + 57600 + off + 0];
    const float by = xyzs[base + 57600 + off + 1];
    const float bz = xyzs[base + 57600 + off + 2];
    sAx[tid] = ax; sAy[tid] = ay; sAz[tid] = az;
    sS4[0 * NPTS + tid] = ax; sS4[1 * NPTS + tid] = ay;
    sS4[2 * NPTS + tid] = az; sS4[3 * NPTS + tid] = 0.0f;
    sB4[0 * NPTS + tid] = bx; sB4[1 * NPTS + tid] = by;
    sB4[2 * NPTS + tid] = bz; sB4[3 * NPTS + tid] = 0.0f;
    sSn[tid] = ax * ax + ay * ay + az * az;
    sBn[tid] = bx * bx + by * by + bz * bz;
  }
  if (tid == 0) { sCtl[0] = 0.0f; sCtl[1] = 0.0f; }
  __syncthreads();

  for (int it = 0; it < MAXIT; ++it) {
    // ---- NN search: cross terms via V_WMMA_F32_16X16X4_F32 ----
#pragma unroll
    for (int r = 0; r < 2; ++r) {
      const int m0 = (2 * wid + r) * 16;
      // A 16x4 frag: lanes0-15 K=0/1, lanes16-31 K=2/3 (K=3 is the zero row).
      v2f afrag;
      afrag.x = sS4[krow + m0 + l16];
      afrag.y = sS4[krow + NPTS + m0 + l16];
      float rn[8];
#pragma unroll
      for (int v = 0; v < 8; ++v) rn[v] = sSn[m0 + v + 8 * half];
      float bv[8]; int bi[8];
#pragma unroll
      for (int v = 0; v < 8; ++v) { bv[v] = 3.402823466e38f; bi[v] = 0; }

      for (int nt = 0; nt < NTILE; ++nt) {
        const int n0 = nt * 16;
        v2f bfrag;  // B 4x16 frag: same K striping, branch-free
        bfrag.x = sB4[krow + n0 + l16];
        bfrag.y = sB4[krow + NPTS + n0 + l16];
        v8f cz = {0.f, 0.f, 0.f, 0.f, 0.f, 0.f, 0.f, 0.f};
        v8f d = __builtin_amdgcn_wmma_f32_16x16x4_f32(
            false, afrag, false, bfrag, (short)0, cz, false, false);
        const float bn = sBn[n0 + l16];
        const int idx = n0 + l16;
#pragma unroll
        for (int v = 0; v < 8; ++v) {
          const float d2 = rn[v] + bn - 2.0f * d[v];
          if (d2 < bv[v]) { bv[v] = d2; bi[v] = idx; }
        }
      }
      // argmin across the 16-lane group (cols), first-occurrence tie-break
#pragma unroll
      for (int v = 0; v < 8; ++v) {
#pragma unroll
        for (int m = 1; m <= 8; m <<= 1) {
          const float ov = __shfl_xor(bv[v], m, 32);
          const int   oi = __shfl_xor(bi[v], m, 32);
          if (ov < bv[v] || (ov == bv[v] && oi < bi[v])) { bv[v] = ov; bi[v] = oi; }
        }
        if (l16 == 0) {
          const int M = m0 + v + 8 * half;
          sNN[M] = bi[v];
          sND[M] = sqrtf(fmaxf(bv[v], 0.0f));
        }
      }
    }
    __syncthreads();

    // ---- fused reduction: centroids, cross-covariance, mean err ----
    {
      const float a0 = sS4[0 * NPTS + tid];
      const float a1 = sS4[1 * NPTS + tid];
      const float a2 = sS4[2 * NPTS + tid];
      const int nn = sNN[tid];
      const float b0 = sB4[0 * NPTS + nn];
      const float b1 = sB4[1 * NPTS + nn];
      const float b2 = sB4[2 * NPTS + nn];
      float vals[16] = {a0, a1, a2, b0, b1, b2,
                        a0 * b0, a0 * b1, a0 * b2,
                        a1 * b0, a1 * b1, a1 * b2,
                        a2 * b0, a2 * b1, a2 * b2,
                        sND[tid]};
#pragma unroll
      for (int k = 0; k < 16; ++k) {
        const float v = wave_sum(vals[k]);
        if (lane == 0) sPart[wid][k] = v;
      }
    }
    __syncthreads();

    if (tid == 0) {
      float s[16];
#pragma unroll
      for (int k = 0; k < 16; ++k) {
        float acc = 0.0f;
        for (int w = 0; w < 6; ++w) acc += sPart[w][k];
        s[k] = acc;
      }
      const float inv = 1.0f / (float)NPTS;
      const float ca[3] = {s[0] * inv, s[1] * inv, s[2] * inv};
      const float cb[3] = {s[3] * inv, s[4] * inv, s[5] * inv};
      float H[3][3];
#pragma unroll
      for (int r = 0; r < 3; ++r)
#pragma unroll
        for (int c = 0; c < 3; ++c)
          H[r][c] = s[6 + r * 3 + c] - (float)NPTS * ca[r] * cb[c];
      float R[3][3], tv[3];
      kabsch3(H, ca, cb, R, tv);
#pragma unroll
      for (int k = 0; k < 9; ++k) sRt[k] = R[k / 3][k % 3];
      sRt[9] = tv[0]; sRt[10] = tv[1]; sRt[11] = tv[2];
      const float mean_err = s[15] * inv;
      sCtl[1] = (fabsf(sCtl[0] - mean_err) < TOLCNV) ? 1.0f : 0.0f;
      sCtl[0] = mean_err;
    }
    __syncthreads();

    // ---- apply transform to src ----
    {
      const float px = sS4[0 * NPTS + tid];
      const float py = sS4[1 * NPTS + tid];
      const float pz = sS4[2 * NPTS + tid];
      const float nx = sRt[0] * px + sRt[1] * py + sRt[2] * pz + sRt[9];
      const float ny = sRt[3] * px + sRt[4] * py + sRt[5] * pz + sRt[10];
      const float nz = sRt[6] * px + sRt[7] * py + sRt[8] * pz + sRt[11];
      sS4[0 * NPTS + tid] = nx; sS4[1 * NPTS + tid] = ny; sS4[2 * NPTS + tid] = nz;
      sSn[tid] = nx * nx + ny * ny + nz * nz;
    }
    __syncthreads();
    if (sCtl[1] != 0.0f) break;   // converged: remaining reference iters are frozen no-ops
  }

  // ---- final fit: original A -> final src (identity correspondence) ----
  {
    const float a0 = sAx[tid], a1 = sAy[tid], a2 = sAz[tid];
    const float b0 = sS4[0 * NPTS + tid];
    const float b1 = sS4[1 * NPTS + tid];
    const float b2 = sS4[2 * NPTS + tid];
    float vals[16] = {a0, a1, a2, b0, b1, b2,
                      a0 * b0, a0 * b1, a0 * b2,
                      a1 * b0, a1 * b1, a1 * b2,
                      a2 * b0, a2 * b1, a2 * b2,
                      0.0f};
#pragma unroll
    for (int k = 0; k < 16; ++k) {
      const float v = wave_sum(vals[k]);
      if (lane == 0) sPart[wid][k] = v;
    }
  }
  __syncthreads();

  if (tid == 0) {
    float s[16];
#pragma unroll
    for (int k = 0; k < 16; ++k) {
      float acc = 0.0f;
      for (int w = 0; w < 6; ++w) acc += sPart[w][k];
      s[k] = acc;
    }
    const float inv = 1.0f / (float)NPTS;
    const float ca[3] = {s[0] * inv, s[1] * inv, s[2] * inv};
    const float cb[3] = {s[3] * inv, s[4] * inv, s[5] * inv};
    float H[3][3];
#pragma unroll
    for (int r = 0; r < 3; ++r)
#pragma unroll
      for (int c = 0; c < 3; ++c)
        H[r][c] = s[6 + r * 3 + c] - (float)NPTS * ca[r] * cb[c];
    float R[3][3], tv[3];
    kabsch3(H, ca, cb, R, tv);

    // pose extraction, matching the reference's axis swaps
    const float tX  = atan2f(R[2][1], R[2][2]);
    const float tYo = atan2f(-R[2][0], sqrtf(R[2][1] * R[2][1] + R[2][2] * R[2][2]));
    const float tZo = atan2f(R[1][0], R[0][0]);
    const float tY = tZo, tZ = tYo;                 // swap as in original forward
    const float theta = atan2f(sqrtf(tX * tX + tY * tY), tZ);
    const float phi   = atan2f(tY, tX);

    float* o = out + (size_t)bidx * 5;
    o[0] = tv[0];   // x = t[0]
    o[1] = tv[2];   // y = t[2]
    o[2] = tv[1];   // z = t[1]
    o[3] = theta;
    o[4] = phi;
  }
}

extern "C" void kernel_launch(void* const* d_in, const int* in_sizes, int n_in,
                              void* d_out, int out_size, void* d_ws, size_t ws_size,
                              hipStream_t stream) {
  (void)n_in; (void)out_size; (void)d_ws; (void)ws_size;
  const float* xyzs = (const float*)d_in[0];
  float* out = (float*)d_out;
  const int nbatch = in_sizes[0] / (2 * 3 * 19200);   // 256
  icp_wmma_kernel<<<nbatch, NPTS, 0, stream>>>(xyzs, out);
}